// GraphCorrelationLayer_b_57260503990851
// MI455X (gfx1250) — compile-verified
//
#include <hip/hip_runtime.h>
#include <hip/hip_bf16.h>

#define NN 768
#define CC 256

typedef float v2f __attribute__((ext_vector_type(2)));
typedef float v8f __attribute__((ext_vector_type(8)));

__device__ __forceinline__ float leaky02(float x) { return fmaxf(x, 0.2f * x); }

// ---------------------------------------------------------------------------
// f32 GEMM via V_WMMA_F32_16X16X4_F32.  D = A[MxK] * B[KxN], row-major.
// One wave (32 lanes) owns one 16x16 output tile; K-loop steps by 4, unrolled
// x4.  act==1 applies ELU at the epilogue (for h_prime), act==0 stores raw.
//
// Per-lane VGPR layouts (CDNA5 ISA 7.12.2):
//   A 16x4 f32 : lane<16 -> row M=lane, a={K0,K1};  lane>=16 -> row M=lane-16, a={K2,K3}
//   B 4x16 f32 : lane<16 -> col N=lane, b={K0 row, K1 row}; lane>=16 -> {K2,K3}
//   C/D 16x16  : vgpr r -> M = r + 8*(lane>=16), N = lane%16
// ---------------------------------------------------------------------------
__global__ __launch_bounds__(256) void wmma_gemm_f32_kernel(
    const float* __restrict__ A, const float* __restrict__ B,
    float* __restrict__ D, int M, int N, int K, int act)
{
    const int lane = threadIdx.x & 31;
    const int waveInBlock = threadIdx.x >> 5;
    const int tilesN = N >> 4;
    const int totalTiles = (M >> 4) * tilesN;
    const int tile = blockIdx.x * 8 + waveInBlock;
    if (tile >= totalTiles) return;          // uniform per wave: EXEC stays all-1s

    const int tm = tile / tilesN;
    const int tn = tile - tm * tilesN;
    const int m0 = tm << 4, n0 = tn << 4;

    const int half = lane >> 4;              // 0 or 1
    const int l16  = lane & 15;

    // a[0] = A[m0+l16][k + 2*half], a[1] = next column (8B-aligned: K even)
    const float* arow = A + (size_t)(m0 + l16) * K + 2 * half;
    // b[0] = B[k + 2*half][n0+l16],  b[1] = B[k + 2*half + 1][n0+l16]
    const float* bcol = B + (size_t)(2 * half) * N + (n0 + l16);

    v8f acc = {};
    for (int k = 0; k < K; k += 16) {
        if (k + 16 < K)  // pull the next B panel toward L2/L0 (global_prefetch_b8)
            __builtin_prefetch(bcol + (size_t)(k + 16) * N, 0, 1);
#pragma unroll
        for (int kk = 0; kk < 16; kk += 4) {
            v2f a = *(const v2f*)(arow + k + kk);
            v2f b;
            b.x = bcol[(size_t)(k + kk) * N];
            b.y = bcol[(size_t)(k + kk + 1) * N];
            // (neg_a, A, neg_b, B, c_mod, C, reuse_a, reuse_b)
            acc = __builtin_amdgcn_wmma_f32_16x16x4_f32(
                false, a, false, b, (short)0, acc, false, false);
        }
    }

#pragma unroll
    for (int r = 0; r < 8; ++r) {
        const int m = m0 + r + 8 * half;
        float v = acc[r];
        if (act == 1) v = (v > 0.f) ? v : (__expf(v) - 1.f);   // ELU
        D[(size_t)m * N + (n0 + l16)] = v;
    }
}

// ---------------------------------------------------------------------------
// Fused scores + row-softmax.  One block per row i (768 blocks, 256 threads =
// 8 waves).  Each wave owns one column j at a time: the 32 lanes split the 256
// channels (8 contiguous channels per lane, two b128 loads), accumulate
//   lin_w[c] * leaky(fc_w0*fi[c] + fc_w1*fj[c] + fc_b)
// lanes 0..2 additionally fold in the 3 coordinate channels, then a wave32
// shfl_xor reduction produces the score.  Softmax (max-sub, exp, sum, scale)
// is done in LDS and the attention row is written to the workspace.
// ---------------------------------------------------------------------------
__global__ __launch_bounds__(256) void scores_softmax_kernel(
    const float* __restrict__ fin,     // [N,C] = f_input
    const float* __restrict__ coords,  // [N,3]
    const float* __restrict__ fc_w, const float* __restrict__ fc_b,
    const float* __restrict__ sc_w, const float* __restrict__ sc_b,
    const float* __restrict__ lin_w, const float* __restrict__ lin_b,
    float* __restrict__ att)           // [N,N]
{
    const int i    = blockIdx.x;
    const int tid  = threadIdx.x;
    const int lane = tid & 31;
    const int wave = tid >> 5;

    __shared__ float s_ai[CC];      // fc_w0 * f_input[i][c] + fc_b
    __shared__ float s_lw[CC];      // lin_w[0..255]
    __shared__ float s_score[NN];
    __shared__ float s_red[256];

    const float fw1 = fc_w[1];
    const float sw0 = sc_w[0], sw1 = sc_w[1], sb = sc_b[0];
    const float lb  = lin_b[0];

    s_ai[tid] = fc_w[0] * fin[(size_t)i * CC + tid] + fc_b[0];
    s_lw[tid] = lin_w[tid];
    __syncthreads();

    // hoist this lane's 8-channel slice into registers (one-time LDS reads)
    float ai[8], lw[8];
#pragma unroll
    for (int u = 0; u < 8; ++u) { ai[u] = s_ai[lane * 8 + u]; lw[u] = s_lw[lane * 8 + u]; }

    float aci = 0.f, lwc = 0.f;
    if (lane < 3) {
        aci = sw0 * coords[(size_t)i * 3 + lane] + sb;
        lwc = lin_w[CC + lane];
    }

    for (int j = wave; j < NN; j += 8) {
        const float4* fj4 = (const float4*)(fin + (size_t)j * CC + lane * 8);
        const float4 f0 = fj4[0], f1 = fj4[1];
        const float fj[8] = { f0.x, f0.y, f0.z, f0.w, f1.x, f1.y, f1.z, f1.w };

        float acc = 0.f;
#pragma unroll
        for (int u = 0; u < 8; ++u)
            acc += lw[u] * leaky02(fmaf(fw1, fj[u], ai[u]));

        if (lane < 3)
            acc += lwc * leaky02(fmaf(sw1, coords[(size_t)j * 3 + lane], aci));

#pragma unroll
        for (int off = 16; off > 0; off >>= 1)
            acc += __shfl_xor(acc, off, 32);

        if (lane == 0)
            s_score[j] = leaky02(acc + lb);
    }
    __syncthreads();

    // row max
    float m = -3.0e38f;
    for (int j = tid; j < NN; j += 256) m = fmaxf(m, s_score[j]);
    s_red[tid] = m;
    __syncthreads();
    for (int s = 128; s > 0; s >>= 1) {
        if (tid < s) s_red[tid] = fmaxf(s_red[tid], s_red[tid + s]);
        __syncthreads();
    }
    const float mx = s_red[0];
    __syncthreads();

    // exp + row sum
    float sum = 0.f;
    for (int j = tid; j < NN; j += 256) {
        const float e = __expf(s_score[j] - mx);
        s_score[j] = e;
        sum += e;
    }
    s_red[tid] = sum;
    __syncthreads();
    for (int s = 128; s > 0; s >>= 1) {
        if (tid < s) s_red[tid] += s_red[tid + s];
        __syncthreads();
    }
    const float inv = 1.f / s_red[0];
    __syncthreads();

    for (int j = tid; j < NN; j += 256)
        att[(size_t)i * NN + j] = s_score[j] * inv;
}

// ---------------------------------------------------------------------------
// Launch: GEMM1 (WMMA f32) -> fused scores/softmax -> GEMM2 (WMMA f32 + ELU).
// Workspace layout: [0, 768*256) f_input ; [768*256, +768*768) attention.
// ---------------------------------------------------------------------------
extern "C" void kernel_launch(void* const* d_in, const int* in_sizes, int n_in,
                              void* d_out, int out_size, void* d_ws, size_t ws_size,
                              hipStream_t stream) {
    const float* features = (const float*)d_in[0];
    const float* coords   = (const float*)d_in[1];
    // d_in[2] = adj, unused by the forward pass
    const float* FC    = (const float*)d_in[3];
    const float* fc_w  = (const float*)d_in[4];
    const float* fc_b  = (const float*)d_in[5];
    const float* sc_w  = (const float*)d_in[6];
    const float* sc_b  = (const float*)d_in[7];
    const float* lin_w = (const float*)d_in[8];
    const float* lin_b = (const float*)d_in[9];
    float* out = (float*)d_out;

    float* ws_f   = (float*)d_ws;            // [768,256] f_input
    float* ws_att = ws_f + NN * CC;          // [768,768] attention

    // 48x16 = 768 output tiles, 8 waves/block -> 96 blocks
    wmma_gemm_f32_kernel<<<96, 256, 0, stream>>>(features, FC, ws_f,
                                                 NN, CC, CC, /*act=*/0);

    scores_softmax_kernel<<<NN, 256, 0, stream>>>(ws_f, coords, fc_w, fc_b,
                                                  sc_w, sc_b, lin_w, lin_b,
                                                  ws_att);

    wmma_gemm_f32_kernel<<<96, 256, 0, stream>>>(ws_att, ws_f, out,
                                                 NN, CC, NN, /*act=*/1);
}